// AttentiondecoderV2_8280696946875
// MI455X (gfx1250) — compile-verified
//
#include <hip/hip_runtime.h>
#include <hip/hip_bf16.h>
#include <math.h>

// Problem constants (match reference)
#define Bn 128
#define Tn 256
#define Hn 1024
#define Vn 32000

typedef float v2f __attribute__((ext_vector_type(2)));
typedef float v8f __attribute__((ext_vector_type(8)));

// ---------------------------------------------------------------------------
// Hardware tanh: gfx1250 has V_TANH_F32 (TRANS op). Prefer the builtin so the
// compiler manages the TRANS co-execution hazard; fall back to inline asm with
// a trailing v_nop (ISA 7.4.3: one independent op / V_NOP after a TRANS op
// before its output is consumed).
// ---------------------------------------------------------------------------
__device__ __forceinline__ float fast_tanh(float x) {
#if __has_builtin(__builtin_amdgcn_tanhf)
    return __builtin_amdgcn_tanhf(x);
#else
    float y;
    asm volatile("v_tanh_f32 %0, %1\n\tv_nop" : "=v"(y) : "v"(x));
    return y;
#endif
}

__device__ __forceinline__ float fast_sigmoid(float x) {
    return 1.0f / (1.0f + __expf(-x));   // __expf -> v_exp_f32
}

// ---------------------------------------------------------------------------
// Kernel 1: scores[b,t] = sum_h tanh(hidden[b,h] + enc[t,b,h]) * vat_w[h] + vat_b
// One 128-thread block per (t,b) pair; coalesced over h.
// ---------------------------------------------------------------------------
__global__ void attn_scores(const float* __restrict__ hidden,
                            const float* __restrict__ enc,
                            const float* __restrict__ vat_w,
                            const float* __restrict__ vat_b,
                            float* __restrict__ scores) {
    const int bid = blockIdx.x;          // T*B blocks
    const int b = bid % Bn;
    const int t = bid / Bn;
    const int tid = threadIdx.x;         // 128 threads

    const float* ep = enc + ((size_t)t * Bn + b) * Hn;
    const float* hp = hidden + (size_t)b * Hn;

    float p = 0.0f;
#pragma unroll 4
    for (int h = tid; h < Hn; h += 128) {
        p += fast_tanh(hp[h] + ep[h]) * vat_w[h];
    }

    __shared__ float red[128];
    red[tid] = p;
    __syncthreads();
    for (int off = 64; off > 0; off >>= 1) {
        if (tid < off) red[tid] += red[tid + off];
        __syncthreads();
    }
    if (tid == 0) scores[(size_t)b * Tn + t] = red[0] + vat_b[0];
}

// ---------------------------------------------------------------------------
// Kernel 2: softmax over T per batch row. One block per b, T==blockDim==256.
// Writes attn_w directly into its d_out slot.
// ---------------------------------------------------------------------------
__global__ void attn_softmax(const float* __restrict__ scores,
                             float* __restrict__ attn) {
    const int b = blockIdx.x;
    const int t = threadIdx.x;           // 256 threads == T
    __shared__ float red[256];

    float s = scores[(size_t)b * Tn + t];
    red[t] = s;
    __syncthreads();
    for (int off = 128; off > 0; off >>= 1) {
        if (t < off) red[t] = fmaxf(red[t], red[t + off]);
        __syncthreads();
    }
    const float mx = red[0];
    __syncthreads();

    const float e = __expf(s - mx);
    red[t] = e;
    __syncthreads();
    for (int off = 128; off > 0; off >>= 1) {
        if (t < off) red[t] += red[t + off];
        __syncthreads();
    }
    attn[(size_t)b * Tn + t] = e / red[0];
}

// ---------------------------------------------------------------------------
// Kernel 3: combined[b, 0:H]   = emb[ids[b], :]
//           combined[b, H:2H]  = sum_t attn_w[b,t] * enc[t,b,:]
// 256 threads/block; each block covers 256 consecutive h of a single b,
// attention row staged in LDS.
// ---------------------------------------------------------------------------
__global__ void build_combined(const int* __restrict__ ids,
                               const float* __restrict__ emb,
                               const float* __restrict__ attn,
                               const float* __restrict__ enc,
                               float* __restrict__ combined) {
    const int idx = blockIdx.x * 256 + threadIdx.x;   // B*H threads
    const int b = idx / Hn;
    const int h = idx % Hn;

    __shared__ float aw[Tn];
    aw[threadIdx.x] = attn[(size_t)b * Tn + threadIdx.x];   // Tn == 256
    __syncthreads();

    float acc = 0.0f;
#pragma unroll 4
    for (int t = 0; t < Tn; ++t) {
        acc += aw[t] * enc[((size_t)t * Bn + b) * Hn + h];
    }

    combined[(size_t)b * (2 * Hn) + h]      = emb[(size_t)ids[b] * Hn + h];
    combined[(size_t)b * (2 * Hn) + Hn + h] = acc;
}

// ---------------------------------------------------------------------------
// Kernel 4: WMMA fp32 GEMM:  C[M,N] = A[M,K] * Bw[N,K]^T + bias (+relu)
// One wave per 16x32 C super-tile (two 16x16 accumulators sharing the A
// fragment: 3 b64 loads feed 2 v_wmma_f32_16x16x4_f32, and the two WMMA
// chains are independent -> no D->A/B RAW hazard stalls).
// Fragment layout per ISA 7.12.2 (f32 16x4 A / 4x16 B):
//   lane L: row = L%16 ; VGPR pair holds K = {0,1} (L<16) or {2,3} (L>=16)
// -> each lane issues one contiguous float2 load per operand per K-step.
// Tile guard is wave-uniform, so EXEC is all-1s in the WMMA path.
// Streaming operands are prefetched one 128B line ahead (global_prefetch_b8).
// ---------------------------------------------------------------------------
__global__ void gemm_bias_act(const float* __restrict__ A,
                              const float* __restrict__ Bw,
                              const float* __restrict__ bias,
                              float* __restrict__ C,
                              int M, int N, int K, int act) {
    const int wave    = blockIdx.x * (blockDim.x >> 5) + (threadIdx.x >> 5);
    const int n2tiles = N >> 5;              // 32-wide super-tiles
    const int tile_m  = wave / n2tiles;
    const int tile_n  = (wave % n2tiles) * 2;
    if (tile_m >= (M >> 4)) return;          // wave-uniform

    const int lane = threadIdx.x & 31;
    const int l16  = lane & 15;
    const int hi   = lane >> 4;              // 0 or 1

    const float* arow  = A  + (size_t)(tile_m * 16 + l16) * K + 2 * hi;
    const float* brow0 = Bw + (size_t)(tile_n * 16 + l16) * K + 2 * hi;
    const float* brow1 = brow0 + (size_t)16 * K;

    v8f acc0 = {0.f, 0.f, 0.f, 0.f, 0.f, 0.f, 0.f, 0.f};
    v8f acc1 = {0.f, 0.f, 0.f, 0.f, 0.f, 0.f, 0.f, 0.f};

    for (int k0 = 0; k0 < K; k0 += 32) {
        if (k0 + 32 < K) {                   // one cacheline ahead
            __builtin_prefetch(arow  + k0 + 32, 0, 1);
            __builtin_prefetch(brow0 + k0 + 32, 0, 1);
            __builtin_prefetch(brow1 + k0 + 32, 0, 1);
        }
#pragma unroll
        for (int ku = 0; ku < 32; ku += 4) {
            const int k = k0 + ku;
            v2f a  = *(const v2f*)(arow  + k);
            v2f b0 = *(const v2f*)(brow0 + k);
            v2f b1 = *(const v2f*)(brow1 + k);
            // 8 args: (neg_a, A, neg_b, B, c_mod, C, reuse_a, reuse_b)
            acc0 = __builtin_amdgcn_wmma_f32_16x16x4_f32(
                false, a, false, b0, (short)0, acc0, false, false);
            acc1 = __builtin_amdgcn_wmma_f32_16x16x4_f32(
                false, a, false, b1, (short)0, acc1, false, false);
        }
    }

    const int n0 = tile_n * 16 + l16;
    const int n1 = n0 + 16;
    const float bv0 = bias ? bias[n0] : 0.0f;
    const float bv1 = bias ? bias[n1] : 0.0f;
#pragma unroll
    for (int r = 0; r < 8; ++r) {
        const int m = tile_m * 16 + r + 8 * hi;
        float v0 = acc0[r] + bv0;
        float v1 = acc1[r] + bv1;
        if (act == 1) { v0 = fmaxf(v0, 0.0f); v1 = fmaxf(v1, 0.0f); }
        C[(size_t)m * N + n0] = v0;
        C[(size_t)m * N + n1] = v1;
    }
}

// ---------------------------------------------------------------------------
// Kernel 5: GRU cell elementwise (PyTorch gate order r,z,n).
// ---------------------------------------------------------------------------
__global__ void gru_cell(const float* __restrict__ gi,
                         const float* __restrict__ gh,
                         const float* __restrict__ hprev,
                         float* __restrict__ hnew) {
    const int idx = blockIdx.x * 256 + threadIdx.x;  // B*H threads
    const int b = idx / Hn;
    const int h = idx % Hn;
    const size_t base = (size_t)b * (3 * Hn) + h;

    const float i_r = gi[base];
    const float i_z = gi[base + Hn];
    const float i_n = gi[base + 2 * Hn];
    const float h_r = gh[base];
    const float h_z = gh[base + Hn];
    const float h_n = gh[base + 2 * Hn];

    const float r = fast_sigmoid(i_r + h_r);
    const float z = fast_sigmoid(i_z + h_z);
    const float n = fast_tanh(i_n + r * h_n);
    hnew[idx] = (1.0f - z) * n + z * hprev[idx];
}

// ---------------------------------------------------------------------------
// Kernel 6: in-place log_softmax over V per row. One block (256 thr) per b.
// ---------------------------------------------------------------------------
__global__ void log_softmax_rows(float* __restrict__ logits) {
    const int b = blockIdx.x;
    const int tid = threadIdx.x;          // 256
    float* row = logits + (size_t)b * Vn;
    __shared__ float red[256];

    float mx = -3.0e38f;
    for (int v = tid; v < Vn; v += 256) mx = fmaxf(mx, row[v]);
    red[tid] = mx;
    __syncthreads();
    for (int off = 128; off > 0; off >>= 1) {
        if (tid < off) red[tid] = fmaxf(red[tid], red[tid + off]);
        __syncthreads();
    }
    mx = red[0];
    __syncthreads();

    float s = 0.0f;
    for (int v = tid; v < Vn; v += 256) s += __expf(row[v] - mx);
    red[tid] = s;
    __syncthreads();
    for (int off = 128; off > 0; off >>= 1) {
        if (tid < off) red[tid] += red[tid + off];
        __syncthreads();
    }
    const float lse = mx + logf(red[0]);

    for (int v = tid; v < Vn; v += 256) row[v] = row[v] - lse;
}

// ---------------------------------------------------------------------------
// Host-side launch
// ---------------------------------------------------------------------------
extern "C" void kernel_launch(void* const* d_in, const int* in_sizes, int n_in,
                              void* d_out, int out_size, void* d_ws, size_t ws_size,
                              hipStream_t stream) {
    const int*   input_ids = (const int*)  d_in[0];
    const float* hidden    = (const float*)d_in[1];   // [1,B,H]
    const float* enc       = (const float*)d_in[2];   // [T,B,H]
    const float* emb       = (const float*)d_in[3];   // [V,H]
    const float* vat_w     = (const float*)d_in[4];   // [1,H]
    const float* vat_b     = (const float*)d_in[5];   // [1]
    const float* comb_w    = (const float*)d_in[6];   // [H,2H]
    const float* comb_b    = (const float*)d_in[7];   // [H]
    const float* gru_w_ih  = (const float*)d_in[8];   // [3H,H]
    const float* gru_w_hh  = (const float*)d_in[9];   // [3H,H]
    const float* gru_b_ih  = (const float*)d_in[10];  // [3H]
    const float* gru_b_hh  = (const float*)d_in[11];  // [3H]
    const float* out_w     = (const float*)d_in[12];  // [V,H]
    const float* out_b     = (const float*)d_in[13];  // [V]

    float* out = (float*)d_out;
    // Output layout (flat, return order): logp [B,V] | h_new [1,B,H] | attn_w [B,1,T]
    float* logp   = out;
    float* h_new  = out + (size_t)Bn * Vn;
    float* attn_w = out + (size_t)Bn * Vn + (size_t)Bn * Hn;

    // Workspace layout (floats)
    float* ws       = (float*)d_ws;
    float* scores   = ws;                                    // B*T
    float* combined = scores   + (size_t)Bn * Tn;            // B*2H
    float* x        = combined + (size_t)Bn * 2 * Hn;        // B*H
    float* gi       = x        + (size_t)Bn * Hn;            // B*3H
    float* gh       = gi       + (size_t)Bn * 3 * Hn;        // B*3H

    // 1) attention scores
    attn_scores<<<Tn * Bn, 128, 0, stream>>>(hidden, enc, vat_w, vat_b, scores);

    // 2) softmax over T (attn_w straight into d_out)
    attn_softmax<<<Bn, Tn, 0, stream>>>(scores, attn_w);

    // 3) embedding gather + attention context -> combined [B,2H]
    build_combined<<<(Bn * Hn) / 256, 256, 0, stream>>>(input_ids, emb, attn_w, enc, combined);

    // 4) x = relu(combined @ comb_w.T + comb_b)   M=128 N=1024 K=2048
    {
        const int waves = (Bn / 16) * (Hn / 32);
        gemm_bias_act<<<waves / 4, 128, 0, stream>>>(combined, comb_w, comb_b, x,
                                                     Bn, Hn, 2 * Hn, 1);
    }
    // 5) gi = x @ gru_w_ih.T + b_ih ; gh = hidden @ gru_w_hh.T + b_hh   N=3072 K=1024
    {
        const int waves = (Bn / 16) * ((3 * Hn) / 32);
        gemm_bias_act<<<waves / 4, 128, 0, stream>>>(x, gru_w_ih, gru_b_ih, gi,
                                                     Bn, 3 * Hn, Hn, 0);
        gemm_bias_act<<<waves / 4, 128, 0, stream>>>(hidden, gru_w_hh, gru_b_hh, gh,
                                                     Bn, 3 * Hn, Hn, 0);
    }

    // 6) GRU cell -> h_new (directly into d_out slot)
    gru_cell<<<(Bn * Hn) / 256, 256, 0, stream>>>(gi, gh, hidden, h_new);

    // 7) logits = h_new @ out_w.T + out_b  (into logp region)  M=128 N=32000 K=1024
    {
        const int waves = (Bn / 16) * (Vn / 32);
        gemm_bias_act<<<waves / 4, 128, 0, stream>>>(h_new, out_w, out_b, logp,
                                                     Bn, Vn, Hn, 0);
    }

    // 8) in-place log_softmax over V
    log_softmax_rows<<<Bn, 256, 0, stream>>>(logp);
}